// ProtoConv2d_45165876085079
// MI455X (gfx1250) — compile-verified
//
#include <hip/hip_runtime.h>
#include <math.h>

typedef __attribute__((ext_vector_type(2))) float v2f;
typedef __attribute__((ext_vector_type(8))) float v8f;

#define B_SZ    32
#define C_IN    64
#define H_IN    56
#define K_CLUST 512
#define PATCH   256
#define OH      57
#define L_OUT   (OH * OH)          // 3249
#define PLANE   (PATCH * L_OUT)    // 831744 per batch in cols layout
#define N_ROWS  (B_SZ * L_OUT)     // 103968 flat rows
#define CO      128

#define A_STRIDE 260               // 16x256 tile, padded (260 % 64 == 4)
#define S_STRIDE 516               // 16x512 tile, padded (516 % 64 == 4)

__device__ __forceinline__ v8f wmma_f32_4(v2f a, v2f b, v8f c) {
    // D = A(16x4 f32) * B(4x16 f32) + C(16x16 f32)
    return __builtin_amdgcn_wmma_f32_16x16x4_f32(
        /*neg_a=*/false, a, /*neg_b=*/false, b,
        /*c_mod=*/(short)0, c, /*reuse_a=*/false, /*reuse_b=*/false);
}

// ---------------------------------------------------------------------------
// Kernel 0: csq[k] = sum_j centers[k][j]^2
// ---------------------------------------------------------------------------
__global__ void k_csq(const float* __restrict__ centers, float* __restrict__ csq) {
    int k = blockIdx.x * blockDim.x + threadIdx.x;
    if (k < K_CLUST) {
        const float* row = centers + k * PATCH;
        float s = 0.f;
        #pragma unroll 8
        for (int j = 0; j < PATCH; ++j) { float v = row[j]; s += v * v; }
        csq[k] = s;
    }
}

// ---------------------------------------------------------------------------
// Kernel 1: fused unfold-gather + dist GEMM + softmax + proto GEMM + blend.
// One 128-thread workgroup handles 16 flat rows end-to-end.
// ---------------------------------------------------------------------------
__global__ void __launch_bounds__(128)
k_proto(const float* __restrict__ x, const float* __restrict__ centers,
        const float* __restrict__ csq, const float* __restrict__ temp_p,
        float* __restrict__ Fout) {
    __shared__ float At[16 * A_STRIDE];   // flat tile [16][256] (padded)
    __shared__ float St[16 * S_STRIDE];   // scores/soft [16][512] (padded)
    __shared__ float red[128];
    __shared__ float xsq[16];
    __shared__ float rmax[16];
    __shared__ float rsumi[16];

    const int tid  = threadIdx.x;
    const int lane = tid & 31;
    const int wave = tid >> 5;
    const long long n0 = (long long)blockIdx.x * 16;
    const float T = *temp_p;

    // ---- stage A tile via implicit unfold gather; partial sum of squares ----
    {
        const int m  = tid >> 3;          // row within tile (0..15)
        const int j0 = (tid & 7) * 32;    // 32 consecutive columns
        const long long base = (n0 + m) * (long long)PATCH;
        float ss = 0.f;
        for (int jj = 0; jj < 32; ++jj) {
            int j = j0 + jj;
            long long idx = base + j;                 // linear index in cols[B,P,L]
            int b   = (int)(idx / PLANE);
            int rem = (int)(idx - (long long)b * PLANE);
            int p   = rem / L_OUT;
            int l   = rem - p * L_OUT;
            int c  = p >> 2, ki = (p >> 1) & 1, kj = p & 1;
            int oy = l / OH, ox = l - oy * OH;
            int iy = oy + ki - 1, ix = ox + kj - 1;   // padded unfold source
            float v = 0.f;
            if ((unsigned)iy < (unsigned)H_IN && (unsigned)ix < (unsigned)H_IN)
                v = x[(((long long)b * C_IN + c) * H_IN + iy) * H_IN + ix];
            At[m * A_STRIDE + j] = v;
            ss += v * v;
        }
        red[tid] = ss;
    }
    __syncthreads();
    if (tid < 16) {
        float s = 0.f;
        #pragma unroll
        for (int q = 0; q < 8; ++q) s += red[tid * 8 + q];
        xsq[tid] = s;
    }
    __syncthreads();

    const int mrow  = lane & 15;   // A: M index / B: N index
    const int khalf = lane >> 4;   // K sub-select

    // ---- GEMM1: S[16x512] = A[16x256] @ centers^T[256x512] ----
    for (int nt = wave; nt < 32; nt += 4) {          // 16-wide center tiles
        v8f acc = {0.f,0.f,0.f,0.f,0.f,0.f,0.f,0.f};
        const float* bptr = centers + (nt * 16 + mrow) * PATCH + khalf * 2;
        const float* aptr = At + mrow * A_STRIDE + khalf * 2;
        #pragma unroll 8
        for (int kb = 0; kb < 64; ++kb) {
            v2f a = *(const v2f*)(aptr + kb * 4);
            v2f b = *(const v2f*)(bptr + kb * 4);
            acc = wmma_f32_4(a, b, acc);
        }
        #pragma unroll
        for (int r = 0; r < 8; ++r) {
            int mm = r + 8 * khalf;                  // D-layout row
            St[mm * S_STRIDE + nt * 16 + mrow] = acc[r];
        }
    }
    __syncthreads();

    // ---- softmax over 512 clusters: logits = -T * sqrt(max(x2+c2-2s,0)) ----
    {
        const int m  = tid >> 3;
        const int nb = (tid & 7) * 64;
        const float xs = xsq[m];
        float lmax = -3.4e38f;
        for (int q = 0; q < 64; ++q) {
            int n = nb + q;
            float s  = St[m * S_STRIDE + n];
            float dd = xs + csq[n] - 2.f * s;
            float lg = -T * sqrtf(fmaxf(dd, 0.f));
            St[m * S_STRIDE + n] = lg;
            lmax = fmaxf(lmax, lg);
        }
        red[tid] = lmax;
        __syncthreads();
        if ((tid & 7) == 0) {
            float mx = red[tid];
            #pragma unroll
            for (int q = 1; q < 8; ++q) mx = fmaxf(mx, red[tid + q]);
            rmax[m] = mx;
        }
        __syncthreads();
        const float rm = rmax[m];
        float lsum = 0.f;
        for (int q = 0; q < 64; ++q) {
            int n = nb + q;
            float e = __expf(St[m * S_STRIDE + n] - rm);
            St[m * S_STRIDE + n] = e;
            lsum += e;
        }
        red[tid] = lsum;
        __syncthreads();
        if ((tid & 7) == 0) {
            float s = 0.f;
            #pragma unroll
            for (int q = 0; q < 8; ++q) s += red[tid + q];
            rsumi[m] = 1.f / s;
        }
        __syncthreads();
        const float ri = rsumi[m];
        for (int q = 0; q < 64; ++q) St[m * S_STRIDE + nb + q] *= ri;
    }
    __syncthreads();

    // ---- GEMM2: Tr[16x256] = soft[16x512] @ centers[512x256]; blend; store ----
    const float invT1 = 1.f / (T + 1.f);
    for (int nt = wave; nt < 16; nt += 4) {          // 16-wide dim tiles
        v8f acc = {0.f,0.f,0.f,0.f,0.f,0.f,0.f,0.f};
        const float* aptr = St + mrow * S_STRIDE + khalf * 2;
        const float* bptr = centers + (khalf * 2) * PATCH + nt * 16 + mrow;
        #pragma unroll 4
        for (int kb = 0; kb < 128; ++kb) {
            v2f a = *(const v2f*)(aptr + kb * 4);
            v2f b;
            b.x = bptr[(kb * 4    ) * PATCH];
            b.y = bptr[(kb * 4 + 1) * PATCH];
            acc = wmma_f32_4(a, b, acc);
        }
        #pragma unroll
        for (int r = 0; r < 8; ++r) {
            int mm = r + 8 * khalf;
            int j  = nt * 16 + mrow;
            float fin = (T * acc[r] + At[mm * A_STRIDE + j]) * invT1;
            Fout[(n0 + mm) * PATCH + j] = fin;       // contiguous cols layout
        }
    }
}

// ---------------------------------------------------------------------------
// Kernel 2: fold(stride=2) + conv(stride=2) collapsed to a masked GEMM:
//   out[b,co,l] = bias[co] + sum_p W[co,p] * mask(p,l) * F[b,p,l]
// One wave owns a 16-wide l-tile and all 128 output channels.
// ---------------------------------------------------------------------------
__global__ void __launch_bounds__(128)
k_foldconv(const float* __restrict__ F, const float* __restrict__ W,
           const float* __restrict__ bias, float* __restrict__ out) {
    const int tid  = threadIdx.x;
    const int lane = tid & 31;
    const int wave = tid >> 5;
    const int b    = blockIdx.x / 51;
    const int lt4  = blockIdx.x - b * 51;
    const int lb   = lt4 * 64 + wave * 16;

    const int ncol  = lane & 15;
    const int khalf = lane >> 4;
    const int l     = lb + ncol;
    const bool lvalid = (l < L_OUT);
    const int ls = lvalid ? l : 0;
    const int oy = ls / OH, ox = ls - oy * OH;

    v8f acc[8];
    #pragma unroll
    for (int i = 0; i < 8; ++i) acc[i] = (v8f){0.f,0.f,0.f,0.f,0.f,0.f,0.f,0.f};

    const float* fbase = F + (long long)b * PLANE + ls;

    for (int kb = 0; kb < 64; ++kb) {
        // B fragment: border-masked F values for k = kb*4 + khalf*2 + {0,1}
        v2f bf;
        #pragma unroll
        for (int r = 0; r < 2; ++r) {
            int k  = kb * 4 + khalf * 2 + r;
            int ki = (k >> 1) & 1, kj = k & 1;
            bool ok = lvalid
                   && !(oy == 0      && ki == 0) && !(oy == OH - 1 && ki == 1)
                   && !(ox == 0      && kj == 0) && !(ox == OH - 1 && kj == 1);
            float v = ok ? fbase[(long long)k * L_OUT] : 0.f;
            if (r == 0) bf.x = v; else bf.y = v;
        }
        #pragma unroll
        for (int ct = 0; ct < 8; ++ct) {
            int co = ct * 16 + ncol;                        // A-matrix M row
            const float* wp = W + co * PATCH + kb * 4 + khalf * 2;
            v2f af = *(const v2f*)wp;
            acc[ct] = wmma_f32_4(af, bf, acc[ct]);
        }
    }

    #pragma unroll
    for (int ct = 0; ct < 8; ++ct) {
        #pragma unroll
        for (int r = 0; r < 8; ++r) {
            int co = ct * 16 + r + 8 * khalf;               // D-layout row
            if (lvalid)
                out[(long long)(b * CO + co) * L_OUT + l] = acc[ct][r] + bias[co];
        }
    }
}

// ---------------------------------------------------------------------------
extern "C" void kernel_launch(void* const* d_in, const int* in_sizes, int n_in,
                              void* d_out, int out_size, void* d_ws, size_t ws_size,
                              hipStream_t stream) {
    const float* x       = (const float*)d_in[0];  // [32,64,56,56]
    const float* weight  = (const float*)d_in[1];  // [128,64,2,2] == [128,256]
    const float* bias    = (const float*)d_in[2];  // [128]
    const float* centers = (const float*)d_in[3];  // [512,256]
    const float* temp    = (const float*)d_in[4];  // scalar
    float* out = (float*)d_out;                    // [32,128,57,57]

    float* csq = (float*)d_ws;                     // 512 floats
    float* F   = (float*)d_ws + 1024;              // [32,256,3249] f32 (~102 MB)

    k_csq<<<2, 256, 0, stream>>>(centers, csq);
    k_proto<<<N_ROWS / 16, 128, 0, stream>>>(x, centers, csq, temp, F);
    k_foldconv<<<B_SZ * 51, 128, 0, stream>>>(F, weight, bias, out);
}